// Affinity_GAT_75557064671579
// MI455X (gfx1250) — compile-verified
//
#include <hip/hip_runtime.h>
#include <hip/hip_bf16.h>

// ---------- vector helpers ----------
typedef float v2f __attribute__((ext_vector_type(2)));
typedef float v8f __attribute__((ext_vector_type(8)));

template <int VEC> struct VecT;
template <> struct VecT<4> { using T = float4; };
template <> struct VecT<2> { using T = float2; };

#define NEG_SLOPE 0.2f

// =====================================================================
// GEMM  C[M,Ncol] = A[M,K] @ B[K,Ncol]   (fp32 WMMA 16x16x4)
// block = (32, Ncol/16); each wave owns one 16x16 tile of C at
// (blockIdx.x*16, threadIdx.y*16).  A tile [16,K] staged in LDS.
// K <= 128, K % 4 == 0.
// =====================================================================
__global__ void gemm_wmma_f32(const float* __restrict__ A,
                              const float* __restrict__ B,
                              float* __restrict__ C,
                              int M, int K, int Ncol) {
  __shared__ float As[16 * 128];

  const int rowBase  = blockIdx.x * 16;
  const int tid      = threadIdx.y * 32 + threadIdx.x;
  const int nthreads = blockDim.x * blockDim.y;

  // cooperative load of A tile (float4 granularity, row-guarded)
  const int totalV4 = (16 * K) / 4;
  for (int i = tid; i < totalV4; i += nthreads) {
    int g   = i * 4;
    int row = rowBase + g / K;
    float4 v = make_float4(0.f, 0.f, 0.f, 0.f);
    if (row < M)
      v = *reinterpret_cast<const float4*>(A + (size_t)row * K + (g % K));
    *reinterpret_cast<float4*>(As + g) = v;
  }
  __syncthreads();

  const int lane = threadIdx.x;
  const int half = lane >> 4;       // 0: lanes 0-15, 1: lanes 16-31
  const int mn   = lane & 15;
  const int col  = threadIdx.y * 16 + mn;

  v8f c = {};
  for (int k0 = 0; k0 < K; k0 += 4) {
    v2f a, b;
    // A 16x4 f32 layout: V0 = K0|K2, V1 = K1|K3 (low|high lane halves)
    a[0] = As[mn * K + k0 + 2 * half];
    a[1] = As[mn * K + k0 + 2 * half + 1];
    // B 4x16 f32 layout: V0 = K0|K2, V1 = K1|K3
    b[0] = B[(size_t)(k0 + 2 * half) * Ncol + col];
    b[1] = B[(size_t)(k0 + 2 * half + 1) * Ncol + col];
    c = __builtin_amdgcn_wmma_f32_16x16x4_f32(false, a, false, b,
                                              (short)0, c, false, false);
  }

  // C/D 16x16 f32: VGPR r -> M=r (lanes 0-15), M=r+8 (lanes 16-31)
#pragma unroll
  for (int r = 0; r < 8; ++r) {
    int row = rowBase + r + 8 * half;
    if (row < M) C[(size_t)row * Ncol + col] = c[r];
  }
}

// =====================================================================
// self-loop attr stats:  attr_sum[d] += ea[e];  cnt[d] += 1
// =====================================================================
__global__ void selfloop_stats(const int* __restrict__ ei,
                               const float* __restrict__ ea,
                               float* __restrict__ attr_sum,
                               float* __restrict__ cntf, int E) {
  int t = blockIdx.x * blockDim.x + threadIdx.x;
  if (t >= E) return;
  int d = ei[E + t];
  atomicAdd(&attr_sum[d], ea[t]);
  atomicAdd(&cntf[d], 1.0f);
}

// build concatenated edge arrays (originals + self loops)
__global__ void build_edges(const int* __restrict__ ei,
                            const float* __restrict__ ea,
                            const float* __restrict__ attr_sum,
                            const float* __restrict__ cntf,
                            int* __restrict__ srcT, int* __restrict__ dstT,
                            float* __restrict__ eaT, int E, int N) {
  int t = blockIdx.x * blockDim.x + threadIdx.x;
  int ET = E + N;
  if (t >= ET) return;
  if (t < E) {
    srcT[t] = ei[t];
    dstT[t] = ei[E + t];
    eaT[t]  = ea[t];
  } else {
    int i = t - E;
    srcT[t] = i;
    dstT[t] = i;
    eaT[t]  = attr_sum[i] / fmaxf(cntf[i], 1.0f);
  }
}

__global__ void deg_kernel(const int* __restrict__ dstT, int* __restrict__ deg,
                           int ET) {
  int t = blockIdx.x * blockDim.x + threadIdx.x;
  if (t >= ET) return;
  atomicAdd(&deg[dstT[t]], 1);
}

// single-workgroup hierarchical exclusive scan (N up to ~1M)
__global__ __launch_bounds__(1024) void scan_kernel(const int* __restrict__ deg,
                                                    int* __restrict__ rowptr,
                                                    int* __restrict__ cursor,
                                                    int n) {
  __shared__ int sums[1024];
  const int T   = 1024;
  const int tid = threadIdx.x;
  const int chunk = (n + T - 1) / T;
  const int beg = tid * chunk;
  const int end = min(beg + chunk, n);

  int s = 0;
  for (int i = beg; i < end; ++i) s += deg[i];
  sums[tid] = s;
  __syncthreads();
  for (int off = 1; off < T; off <<= 1) {
    int t = (tid >= off) ? sums[tid - off] : 0;
    __syncthreads();
    sums[tid] += t;
    __syncthreads();
  }
  int run = sums[tid] - s;  // exclusive prefix for this thread's chunk
  for (int i = beg; i < end; ++i) {
    rowptr[i] = run;
    cursor[i] = run;
    run += deg[i];
  }
  if (tid == T - 1) rowptr[n] = sums[T - 1];
}

__global__ void fill_csr(const int* __restrict__ dstT, int* __restrict__ cursor,
                         int* __restrict__ eidb, int ET) {
  int t = blockIdx.x * blockDim.x + threadIdx.x;
  if (t >= ET) return;
  int pos = atomicAdd(&cursor[dstT[t]], 1);
  eidb[pos] = t;
}

// =====================================================================
// per-edge logits: l_e = att . leaky_relu(xl[src] + xr[dst] + ea*We)
// one wave32 per edge; lane owns VEC contiguous features (C = 32*VEC)
// =====================================================================
template <int VEC>
__global__ void edge_logits_kernel(const float* __restrict__ xl,
                                   const float* __restrict__ xr,
                                   const int* __restrict__ srcT,
                                   const int* __restrict__ dstT,
                                   const float* __restrict__ eaT,
                                   const float* __restrict__ We,
                                   const float* __restrict__ att,
                                   float* __restrict__ logits, int ET) {
  const int wid  = (blockIdx.x * blockDim.x + threadIdx.x) >> 5;
  const int lane = threadIdx.x & 31;
  if (wid >= ET) return;
  const int C = 32 * VEC;
  const int s = srcT[wid], d = dstT[wid];
  const float ea = eaT[wid];

  using VT = typename VecT<VEC>::T;
  float xa[VEC], xb[VEC];
  *reinterpret_cast<VT*>(xa) =
      *reinterpret_cast<const VT*>(xl + (size_t)s * C + lane * VEC);
  *reinterpret_cast<VT*>(xb) =
      *reinterpret_cast<const VT*>(xr + (size_t)d * C + lane * VEC);

  float part = 0.f;
#pragma unroll
  for (int v = 0; v < VEC; ++v) {
    float z = xa[v] + xb[v] + ea * We[lane * VEC + v];
    z = (z > 0.f) ? z : NEG_SLOPE * z;
    part += z * att[lane * VEC + v];
  }
#pragma unroll
  for (int off = 16; off; off >>= 1) part += __shfl_xor(part, off, 32);
  if (lane == 0) logits[wid] = part;
}

// =====================================================================
// per-node segment softmax + weighted aggregation + bias + ELU
// one wave32 per node; CSR (rowptr, eidb) gives incoming edges
// =====================================================================
template <int VEC>
__global__ void node_aggregate_kernel(const float* __restrict__ xl,
                                      const float* __restrict__ logits,
                                      const int* __restrict__ eidb,
                                      const int* __restrict__ rowptr,
                                      const int* __restrict__ srcT,
                                      const float* __restrict__ bias,
                                      float* __restrict__ out, int N) {
  const int node = (blockIdx.x * blockDim.x + threadIdx.x) >> 5;
  const int lane = threadIdx.x & 31;
  if (node >= N) return;
  const int C   = 32 * VEC;
  const int beg = rowptr[node], end = rowptr[node + 1];

  // segment max
  float m = -3.4e38f;
  for (int j = beg + lane; j < end; j += 32) m = fmaxf(m, logits[eidb[j]]);
#pragma unroll
  for (int off = 16; off; off >>= 1) m = fmaxf(m, __shfl_xor(m, off, 32));

  // segment sum of exp
  float ssum = 0.f;
  for (int j = beg + lane; j < end; j += 32)
    ssum += __expf(logits[eidb[j]] - m);
#pragma unroll
  for (int off = 16; off; off >>= 1) ssum += __shfl_xor(ssum, off, 32);
  const float inv = 1.0f / (ssum + 1e-16f);

  // weighted accumulation (all lanes walk edges together; lane owns VEC feats)
  using VT = typename VecT<VEC>::T;
  float acc[VEC];
#pragma unroll
  for (int v = 0; v < VEC; ++v) acc[v] = 0.f;

  for (int j = beg; j < end; ++j) {
    int e = eidb[j];
    float w = __expf(logits[e] - m) * inv;
    float row[VEC];
    *reinterpret_cast<VT*>(row) = *reinterpret_cast<const VT*>(
        xl + (size_t)srcT[e] * C + lane * VEC);
#pragma unroll
    for (int v = 0; v < VEC; ++v) acc[v] += w * row[v];
  }

#pragma unroll
  for (int v = 0; v < VEC; ++v) {
    float o = acc[v] + bias[lane * VEC + v];
    out[(size_t)node * C + lane * VEC + v] =
        (o > 0.f) ? o : (__expf(o) - 1.0f);
  }
}

// =====================================================================
// host launcher
// =====================================================================
extern "C" void kernel_launch(void* const* d_in, const int* in_sizes, int n_in,
                              void* d_out, int out_size, void* d_ws,
                              size_t ws_size, hipStream_t stream) {
  const float* x   = (const float*)d_in[0];
  const int*   ei  = (const int*)d_in[1];
  const float* ea  = (const float*)d_in[2];
  const float* Wl1 = (const float*)d_in[3];
  const float* Wr1 = (const float*)d_in[4];
  const float* We1 = (const float*)d_in[5];
  const float* at1 = (const float*)d_in[6];
  const float* b1  = (const float*)d_in[7];
  const float* Wl2 = (const float*)d_in[8];
  const float* Wr2 = (const float*)d_in[9];
  const float* We2 = (const float*)d_in[10];
  const float* at2 = (const float*)d_in[11];
  const float* b2  = (const float*)d_in[12];
  float* out = (float*)d_out;

  const int H    = in_sizes[6];        // 128
  const int OUT  = in_sizes[11];       // 64
  const int F_IN = in_sizes[3] / H;    // 128
  const int N    = in_sizes[0] / F_IN; // 50000
  const int E    = in_sizes[1] / 2;    // 800000
  const int ET   = E + N;              // 850000

  // ---- workspace carve-out ----
  char* base = (char*)d_ws;
  size_t off = 0;
  auto carve = [&](size_t bytes) -> char* {
    char* p = base + off;
    off += (bytes + 255) & ~size_t(255);
    return p;
  };
  float* xl1      = (float*)carve((size_t)N * H * sizeof(float));
  float* xr1      = (float*)carve((size_t)N * H * sizeof(float));
  float* h        = (float*)carve((size_t)N * H * sizeof(float));
  float* logitsb  = (float*)carve((size_t)ET * sizeof(float));
  int*   srcT     = (int*)carve((size_t)ET * sizeof(int));
  int*   dstT     = (int*)carve((size_t)ET * sizeof(int));
  float* eaT      = (float*)carve((size_t)ET * sizeof(float));
  int*   eidb     = (int*)carve((size_t)ET * sizeof(int));
  int*   rowptr   = (int*)carve((size_t)(N + 1) * sizeof(int));
  int*   cursor   = (int*)carve((size_t)N * sizeof(int));
  int*   deg      = (int*)carve((size_t)N * sizeof(int));
  float* attr_sum = (float*)carve((size_t)N * sizeof(float));
  float* cntf     = (float*)carve((size_t)N * sizeof(float));
  // layer-2 transforms reuse xr1's space (N*H == 2*N*OUT)
  float* xl2 = xr1;
  float* xr2 = xr1 + (size_t)N * OUT;
  (void)ws_size; (void)n_in; (void)out_size;

  const int TB = 256;
  auto gb = [](int n, int t) { return (unsigned)((n + t - 1) / t); };

  // ---- graph prep (once, reused by both layers) ----
  hipMemsetAsync(attr_sum, 0, (size_t)N * sizeof(float), stream);
  hipMemsetAsync(cntf, 0, (size_t)N * sizeof(float), stream);
  hipMemsetAsync(deg, 0, (size_t)N * sizeof(int), stream);

  selfloop_stats<<<gb(E, TB), TB, 0, stream>>>(ei, ea, attr_sum, cntf, E);
  build_edges<<<gb(ET, TB), TB, 0, stream>>>(ei, ea, attr_sum, cntf, srcT,
                                             dstT, eaT, E, N);
  deg_kernel<<<gb(ET, TB), TB, 0, stream>>>(dstT, deg, ET);
  scan_kernel<<<1, 1024, 0, stream>>>(deg, rowptr, cursor, N);
  fill_csr<<<gb(ET, TB), TB, 0, stream>>>(dstT, cursor, eidb, ET);

  // ---- layer 1 (C = H = 128) ----
  {
    dim3 blk(32, H / 16);
    gemm_wmma_f32<<<gb(N, 16), blk, 0, stream>>>(x, Wl1, xl1, N, F_IN, H);
    gemm_wmma_f32<<<gb(N, 16), blk, 0, stream>>>(x, Wr1, xr1, N, F_IN, H);
  }
  edge_logits_kernel<4><<<gb(ET * 32, TB), TB, 0, stream>>>(
      xl1, xr1, srcT, dstT, eaT, We1, at1, logitsb, ET);
  node_aggregate_kernel<4><<<gb(N * 32, TB), TB, 0, stream>>>(
      xl1, logitsb, eidb, rowptr, srcT, b1, h, N);

  // ---- layer 2 (C = OUT = 64) ----
  {
    dim3 blk(32, OUT / 16);
    gemm_wmma_f32<<<gb(N, 16), blk, 0, stream>>>(h, Wl2, xl2, N, H, OUT);
    gemm_wmma_f32<<<gb(N, 16), blk, 0, stream>>>(h, Wr2, xr2, N, H, OUT);
  }
  edge_logits_kernel<2><<<gb(ET * 32, TB), TB, 0, stream>>>(
      xl2, xr2, srcT, dstT, eaT, We2, at2, logitsb, ET);
  node_aggregate_kernel<2><<<gb(N * 32, TB), TB, 0, stream>>>(
      xl2, logitsb, eidb, rowptr, srcT, b2, out, N);
}